// ConvKB_3504693314221
// MI455X (gfx1250) — compile-verified
//
#include <hip/hip_runtime.h>

// Non-precise FP at file scope: gives nnan/nsz fast-math flags to expressions
// in THIS file. Combined with calling __builtin_fmaxf directly (instead of the
// header-inline fmaxf, whose llvm.maxnum carries the header's FP state), the
// backend emits a single v_max_num_f32 per ReLU instead of canonicalize+max.
#pragma float_control(precise, off)

// ConvKB scoring on MI455X (gfx1250), wave32 + V_WMMA_F32_16X16X4_F32.
//
// score[e] = sum_{c,d} relu( cw[c,0]*h[row[e],d] + cw[c,1]*h[col[e],d]
//                          + cw[c,2]*g[type[e],d] + cb[c] ) * lw[c*D+d] + lb
//
// One wave handles 16 edges. For each d, one WMMA computes y[16 edges][16 ch]
// = A(16x4) x B(4x16) + bias; two WMMAs cover C=32 channels. ReLU + weighted
// accumulate stays in VALU registers (co-executes with the XDL pipe); a
// 16-lane shfl_xor tree reduces the channel dimension at the end.

typedef float v2f __attribute__((ext_vector_type(2)));
typedef float v8f __attribute__((ext_vector_type(8)));
typedef float f4v __attribute__((ext_vector_type(4)));

#define DIM   128
#define NCH   32
#define NEDGE 50000
#define NBLK  (NEDGE / 16)   // 3125 (E divisible by 16)
#define WAVES_PER_WG 8

__global__ __launch_bounds__(256) void convkb_wmma_kernel(
    const float* __restrict__ h,      // [N_NODES, 128]
    const float* __restrict__ g,      // [N_REL, 128]
    const int*   __restrict__ eidx,   // [2, E] flat: row=eidx[0:E], col=eidx[E:2E]
    const int*   __restrict__ etype,  // [E]
    const float* __restrict__ cw,     // [32, 3]
    const float* __restrict__ cb,     // [32]
    const float* __restrict__ lw,     // [32*128], index c*128+d
    const float* __restrict__ lb,     // [1]
    float*       __restrict__ out)    // [E]
{
    const int lane = threadIdx.x & 31;
    const int wave = threadIdx.x >> 5;
    const int blk  = blockIdx.x * WAVES_PER_WG + wave;
    if (blk >= NBLK) return;          // uniform per-wave: EXEC stays all-ones

    const int  n  = lane & 15;        // WMMA column (channel) / A row (edge)
    const bool lo = lane < 16;

    // ---- B operands (constant): conv_w^T padded to 4x16, two channel halves.
    // f32 B 4x16 layout: N = lane&15; VGPR0 = K0 (lanes 0-15) / K2 (lanes 16-31),
    // VGPR1 = K1 / K3 (K3 is the zero pad).
    v2f b0, b1;
    {
        const int ch0 = n, ch1 = n + 16;
        b0.x = lo ? cw[ch0 * 3 + 0] : cw[ch0 * 3 + 2];
        b0.y = lo ? cw[ch0 * 3 + 1] : 0.0f;
        b1.x = lo ? cw[ch1 * 3 + 0] : cw[ch1 * 3 + 2];
        b1.y = lo ? cw[ch1 * 3 + 1] : 0.0f;
    }

    // ---- C operand: bias splat. C/D layout: column N = lane&15 for both lane
    // halves, so every D-VGPR of a lane shares one channel -> one splat value.
    v8f cbias0, cbias1;
    {
        const float bv0 = cb[n], bv1 = cb[n + 16];
        #pragma unroll
        for (int v = 0; v < 8; ++v) { cbias0[v] = bv0; cbias1[v] = bv1; }
    }

    // ---- Per-lane A-source pointers. A 16x4 f32 layout: M = lane&15 for both
    // halves; lanes 0-15 carry K0/K1 (head/tail), lanes 16-31 carry K2/K3
    // (relation / zero pad).
    const int e = blk * 16 + n;
    const int r = eidx[e];
    const int c = eidx[NEDGE + e];
    const int t = etype[e];
    const float* pA0 = lo ? (h + (size_t)r * DIM) : (g + (size_t)t * DIM);
    const float* pA1 = lo ? (h + (size_t)c * DIM) : pA0;  // hi-lane value discarded
    const float* plw0 = lw + (size_t)n * DIM;             // channels 0-15
    const float* plw1 = lw + (size_t)(n + 16) * DIM;      // channels 16-31

    v8f acc;
    #pragma unroll
    for (int v = 0; v < 8; ++v) acc[v] = 0.0f;

    for (int d = 0; d < DIM; d += 4) {
        const f4v xa = *(const f4v*)(pA0 + d);   // lo: h[row]  hi: g[type]
        const f4v xb = *(const f4v*)(pA1 + d);   // lo: h[col]  hi: dummy
        const f4v w0 = *(const f4v*)(plw0 + d);
        const f4v w1 = *(const f4v*)(plw1 + d);
        #pragma unroll
        for (int j = 0; j < 4; ++j) {
            v2f a;
            a.x = xa[j];
            a.y = lo ? xb[j] : 0.0f;             // K3 zero pad in hi lanes
            v8f y0 = __builtin_amdgcn_wmma_f32_16x16x4_f32(
                false, a, false, b0, (short)0, cbias0, false, false);
            v8f y1 = __builtin_amdgcn_wmma_f32_16x16x4_f32(
                false, a, false, b1, (short)0, cbias1, false, false);
            #pragma unroll
            for (int v = 0; v < 8; ++v) {
                // __builtin_fmaxf at this call site carries the pragma's
                // nnan/nsz flags -> single v_max_num_f32, no canonicalize.
                acc[v] = __builtin_fmaf(__builtin_fmaxf(y0[v], 0.0f), w0[j], acc[v]);
                acc[v] = __builtin_fmaf(__builtin_fmaxf(y1[v], 0.0f), w1[j], acc[v]);
            }
        }
    }

    // ---- Reduce over channels: D layout puts edge m=v (lanes 0-15) and
    // m=v+8 (lanes 16-31); xor masks < 16 stay inside each 16-lane half.
    #pragma unroll
    for (int v = 0; v < 8; ++v) {
        float s = acc[v];
        #pragma unroll
        for (int m = 1; m < 16; m <<= 1) s += __shfl_xor(s, m, 32);
        acc[v] = s;
    }

    if (n == 0) {                      // lane 0 -> edges 0-7, lane 16 -> 8-15
        const float bias = lb[0];
        const int base = blk * 16 + (lo ? 0 : 8);
        #pragma unroll
        for (int v = 0; v < 8; ++v) out[base + v] = acc[v] + bias;
    }
}

extern "C" void kernel_launch(void* const* d_in, const int* in_sizes, int n_in,
                              void* d_out, int out_size, void* d_ws, size_t ws_size,
                              hipStream_t stream) {
    const float* h     = (const float*)d_in[0];
    const float* g     = (const float*)d_in[1];
    const int*   eidx  = (const int*)d_in[2];
    const int*   etype = (const int*)d_in[3];
    const float* cw    = (const float*)d_in[4];
    const float* cb    = (const float*)d_in[5];
    const float* lwv   = (const float*)d_in[6];
    const float* lbv   = (const float*)d_in[7];
    float* out = (float*)d_out;

    const int grid = (NBLK + WAVES_PER_WG - 1) / WAVES_PER_WG;  // 391
    convkb_wmma_kernel<<<grid, 32 * WAVES_PER_WG, 0, stream>>>(
        h, g, eidx, etype, cw, cb, lwv, lbv, out);
}